// GPN_Encoder_38560216384246
// MI455X (gfx1250) — compile-verified
//
#include <hip/hip_runtime.h>

typedef __attribute__((ext_vector_type(2))) float v2f;
typedef __attribute__((ext_vector_type(8))) float v8f;

// ---------------------------------------------------------------------------
// Fill buf[N*D] with bias[d] (d = i & (D-1)); D is a power of two (128 / 64).
// Used to pre-seed the aggregation targets with the layer bias so that the
// scatter-add produces agg + b directly.
// ---------------------------------------------------------------------------
__global__ void init_bias_kernel(float* __restrict__ buf,
                                 const float* __restrict__ bias,
                                 int total, int dMask) {
    int i = blockIdx.x * blockDim.x + threadIdx.x;
    if (i < total) buf[i] = bias[i & dMask];
}

// ---------------------------------------------------------------------------
// support[N x Dout] = act(A[N x K]) @ W[K x Dout]   (f32, exact)
// One wave32 per 16x16 output tile using V_WMMA_F32_16X16X4_F32 (K-step 4).
// N is a multiple of 16 so EXEC is all-ones in every active wave (WMMA req).
//
// A-operand (16x4 f32): lane L -> row M = L&15; float2 at K = k0 + 2*(L>>4).
// B-operand (4x16 f32): lane L -> col N = L&15; rows k0+2*(L>>4), +1.
// C/D (16x16 f32): VGPR j -> row tile+8*(L>>4)+j, col L&15.
// ---------------------------------------------------------------------------
template <bool RELU_A>
__global__ void gemm_wmma_f32_kernel(const float* __restrict__ A,
                                     const float* __restrict__ W,
                                     float* __restrict__ out,
                                     int N, int K, int Dout) {
    const int lane   = threadIdx.x & 31;
    const int wave   = threadIdx.x >> 5;
    const int waves  = blockDim.x >> 5;
    const int tileId = blockIdx.x * waves + wave;

    const int tilesPerRow = Dout >> 4;
    const int mTile = tileId / tilesPerRow;
    const int nTile = tileId - mTile * tilesPerRow;
    if (mTile * 16 >= N) return;           // wave-uniform: EXEC stays all-ones

    const int lo  = lane & 15;
    const int hi  = lane >> 4;             // 0 or 1
    const int col = nTile * 16 + lo;

    const float* arow = A + (size_t)(mTile * 16 + lo) * K;
    const float* wcol = W + col;

    v8f c = {};
    for (int k0 = 0; k0 < K; k0 += 4) {
        const int ka = k0 + 2 * hi;        // even -> 8B-aligned float2 load
        v2f a;
        {
            const float2 av = *(const float2*)(arow + ka);
            a.x = av.x; a.y = av.y;
        }
        if (RELU_A) {
            a.x = fmaxf(a.x, 0.0f);
            a.y = fmaxf(a.y, 0.0f);
        }
        v2f b;
        b.x = wcol[(size_t)ka * Dout];
        b.y = wcol[(size_t)(ka + 1) * Dout];
        c = __builtin_amdgcn_wmma_f32_16x16x4_f32(
                /*neg_a=*/false, a, /*neg_b=*/false, b,
                /*c_mod=*/(short)0, c, /*reuse_a=*/false, /*reuse_b=*/false);
    }

    float* orow = out + (size_t)(mTile * 16 + hi * 8) * Dout + col;
#pragma unroll
    for (int j = 0; j < 8; ++j) {
        orow[(size_t)j * Dout] = c[j];
    }
}

// ---------------------------------------------------------------------------
// agg[dst[e]] += w[e] * support[src[e]]   (row-wise, D = 32*FPL floats)
// One wave32 per edge; each lane owns FPL consecutive floats (float4/float2
// vector gather, FPL scalar f32 atomics). Working set lives in the 192MB L2.
// ---------------------------------------------------------------------------
template <int FPL>
__global__ void scatter_add_kernel(const float* __restrict__ support,
                                   const int* __restrict__ esrc,
                                   const int* __restrict__ edst,
                                   const float* __restrict__ ew,
                                   float* __restrict__ agg,
                                   int nEdges) {
    const int D = 32 * FPL;
    const int e = (blockIdx.x * blockDim.x + threadIdx.x) >> 5;
    if (e >= nEdges) return;
    const int lane = threadIdx.x & 31;

    const float wgt = ew[e];
    const float* srow = support + (size_t)esrc[e] * D + lane * FPL;
    float* drow       = agg     + (size_t)edst[e] * D + lane * FPL;

    if (FPL == 4) {
        const float4 v = *(const float4*)srow;
        atomicAdd(drow + 0, wgt * v.x);
        atomicAdd(drow + 1, wgt * v.y);
        atomicAdd(drow + 2, wgt * v.z);
        atomicAdd(drow + 3, wgt * v.w);
    } else {
        const float2 v = *(const float2*)srow;
        atomicAdd(drow + 0, wgt * v.x);
        atomicAdd(drow + 1, wgt * v.y);
    }
}

// ---------------------------------------------------------------------------
// Launch: x, edge_src, edge_dst, edge_weight, W1, b1, W2, b2
// ---------------------------------------------------------------------------
extern "C" void kernel_launch(void* const* d_in, const int* in_sizes, int n_in,
                              void* d_out, int out_size, void* d_ws, size_t ws_size,
                              hipStream_t stream) {
    const float* x    = (const float*)d_in[0];
    const int*   esrc = (const int*)  d_in[1];
    const int*   edst = (const int*)  d_in[2];
    const float* ew   = (const float*)d_in[3];
    const float* W1   = (const float*)d_in[4];
    const float* b1   = (const float*)d_in[5];
    const float* W2   = (const float*)d_in[6];
    const float* b2   = (const float*)d_in[7];
    float*       out  = (float*)d_out;

    const int NFEAT = 128;
    const int NHID2 = 128;   // 2*NHID
    const int NHID  = 64;
    const int N = in_sizes[0] / NFEAT;   // 40000
    const int E = in_sizes[1];           // 640000

    float* support1 = (float*)d_ws;                      // [N,128]
    float* agg1     = support1 + (size_t)N * NHID2;      // [N,128]
    float* support2 = agg1     + (size_t)N * NHID2;      // [N,64]

    const int BLK = 256;

    // 1) agg1 = b1 (broadcast); d_out = b2 (broadcast)
    {
        int total1 = N * NHID2;
        init_bias_kernel<<<(total1 + BLK - 1) / BLK, BLK, 0, stream>>>(
            agg1, b1, total1, NHID2 - 1);
        int total2 = N * NHID;
        init_bias_kernel<<<(total2 + BLK - 1) / BLK, BLK, 0, stream>>>(
            out, b2, total2, NHID - 1);
    }

    // 2) support1 = x @ W1   (WMMA f32)
    {
        int tiles  = (N / 16) * (NHID2 / 16);
        int blocks = (tiles * 32 + BLK - 1) / BLK;
        gemm_wmma_f32_kernel<false><<<blocks, BLK, 0, stream>>>(
            x, W1, support1, N, NFEAT, NHID2);
    }

    // 3) agg1 += scatter(w * support1[src] -> dst)
    {
        int blocks = ((E * 32) + BLK - 1) / BLK;
        scatter_add_kernel<4><<<blocks, BLK, 0, stream>>>(
            support1, esrc, edst, ew, agg1, E);
    }

    // 4) support2 = relu(agg1) @ W2   (bias b1 already inside agg1)
    {
        int tiles  = (N / 16) * (NHID / 16);
        int blocks = (tiles * 32 + BLK - 1) / BLK;
        gemm_wmma_f32_kernel<true><<<blocks, BLK, 0, stream>>>(
            agg1, W2, support2, N, NHID2, NHID);
    }

    // 5) out(=b2) += scatter(w * support2[src] -> dst)
    {
        int blocks = ((E * 32) + BLK - 1) / BLK;
        scatter_add_kernel<2><<<blocks, BLK, 0, stream>>>(
            support2, esrc, edst, ew, out, E);
    }
}